// WFRFlexPointCloudLoss_44444321579530
// MI455X (gfx1250) — compile-verified
//
#include <hip/hip_runtime.h>

// ---------------------------------------------------------------------------
// WFR (unbalanced) Sinkhorn for MI455X / gfx1250.
// K [8,1024,1024] fp32 is materialized directly in d_out (+24 floats) and is
// L2-resident (32MB << 192MB).  The two per-iteration reductions use
// V_WMMA_F32_16X16X4_F32; the row pass stages transposed A-fragments through
// padded LDS; absorption is a device-side flag (graph-capture safe).
// ---------------------------------------------------------------------------

namespace {
constexpr int   B_     = 8;
constexpr int   I_     = 1024;
constexpr int   J_     = 1024;
constexpr int   NITER_ = 50;
constexpr float EPS_   = 0.1f;
constexpr float PCOEF_ = 1.0f / 1.1f;   // 1/(1+eps)
constexpr float HUGE_  = 1e30f;
constexpr float BIG_   = 1e20f;
constexpr float SMALL_ = 1e-7f;
constexpr float DX_    = 1.0f / (float)I_;
constexpr float DY_    = 1.0f / (float)J_;
}

typedef float v2f __attribute__((ext_vector_type(2)));
typedef float v8f __attribute__((ext_vector_type(8)));

// ---------------------------------------------------------------------------
// init: u = v = 0, b = 1
__global__ __launch_bounds__(256)
void k_init(float* __restrict__ u, float* __restrict__ v, float* __restrict__ bvec)
{
    int t = blockIdx.x * blockDim.x + threadIdx.x;
    if (t < B_ * I_) { u[t] = 0.0f; v[t] = 0.0f; bvec[t] = 1.0f; }
}

// ---------------------------------------------------------------------------
// K = clip(exp((-C + u_i + v_j)/eps), 0, HUGE); predicated on device flag.
__global__ __launch_bounds__(256)
void k_recompute(const float* __restrict__ C, const float* __restrict__ u,
                 const float* __restrict__ v, float* __restrict__ K,
                 const int* __restrict__ flag, int force)
{
    if (!force && *flag == 0) return;
    const size_t t   = (size_t)blockIdx.x * blockDim.x + threadIdx.x;
    const size_t e   = t * 4;                       // one float4 per thread
    const int    b   = (int)(e >> 20);              // / (1024*1024)
    const int    rem = (int)(e & (size_t)(I_ * J_ - 1));
    const int    i   = rem >> 10;
    const int    j   = rem & (J_ - 1);
    const float  ub  = u[b * I_ + i];
    const float4 c4  = *(const float4*)(C + e);
    const float4 v4  = *(const float4*)(v + b * J_ + j);
    const float  r   = 1.0f / EPS_;
    float4 k4;
    k4.x = fminf(__expf((-c4.x + ub + v4.x) * r), HUGE_);
    k4.y = fminf(__expf((-c4.y + ub + v4.y) * r), HUGE_);
    k4.z = fminf(__expf((-c4.z + ub + v4.z) * r), HUGE_);
    k4.w = fminf(__expf((-c4.w + ub + v4.w) * r), HUGE_);
    *(float4*)(K + e) = k4;
}

// ---------------------------------------------------------------------------
// Row pass: s_i = scale * sum_j K_ij * b_j ; optional a_i = (mu_i/(e^u_i s_i))^p
// Each wave owns 16 rows x all 1024 cols.  K tile -> LDS (pad 36) -> A-frag;
// b broadcast -> B-frag; D rows replicate the row sums.
__global__ __launch_bounds__(256)
void k_rowpass(const float* __restrict__ K, const float* __restrict__ bvec,
               float* __restrict__ s_out, const float* __restrict__ marg,
               const float* __restrict__ uvec, float* __restrict__ a_out,
               float scale, int compute_a)
{
    __shared__ float tile[8][16 * 36];            // 8 waves, 16x32 tile, pad 36
    const int wave = threadIdx.x >> 5;
    const int lane = threadIdx.x & 31;
    const int b    = blockIdx.x >> 3;             // 8 row-strips per batch
    const int i0   = (blockIdx.x & 7) * 128 + wave * 16;
    const float* Kb = K + (size_t)b * I_ * J_;
    const float* bb = bvec + (size_t)b * J_;

    const int row = lane & 15;                    // A-frag: row per lane
    const int khi = lane >> 4;                    // 0 -> K={0,1}, 1 -> K={2,3}

    v8f acc = {};
    for (int j0 = 0; j0 < J_; j0 += 32) {
        // stage this wave's 16x32 K tile, coalesced float2 rows
        #pragma unroll
        for (int r = 0; r < 16; ++r) {
            v2f d = *(const v2f*)(Kb + (size_t)(i0 + r) * J_ + j0 + 2 * lane);
            tile[wave][r * 36 + 2 * lane]     = d.x;
            tile[wave][r * 36 + 2 * lane + 1] = d.y;
        }
        if (j0 + 32 < J_)
            __builtin_prefetch(Kb + (size_t)i0 * J_ + j0 + 32, 0, 0);
        // (compiler inserts s_wait_dscnt for the LDS store->load dependence;
        //  tile is private to this wave so no barrier is required)
        #pragma unroll
        for (int jj = 0; jj < 32; jj += 4) {
            v2f afrag;                            // A[row][2*khi], A[row][2*khi+1]
            afrag.x = tile[wave][row * 36 + jj + 2 * khi];
            afrag.y = tile[wave][row * 36 + jj + 2 * khi + 1];
            v2f bfrag;                            // B[k][n] = b[j0+jj+k] broadcast
            bfrag.x = bb[j0 + jj + khi];
            bfrag.y = bb[j0 + jj + 2 + khi];
            acc = __builtin_amdgcn_wmma_f32_16x16x4_f32(false, afrag, false, bfrag,
                                                        (short)0, acc, false, false);
        }
    }
    // D layout: vgpr r, lanes0-15 -> M=r, lanes16-31 -> M=r+8 (rowsum in every N)
    if ((lane & 15) == 0) {
        const int base = i0 + 8 * khi;
        #pragma unroll
        for (int r = 0; r < 8; ++r) {
            const float sv = acc[r] * scale;
            s_out[(size_t)b * I_ + base + r] = sv;
            if (compute_a) {
                const float x  = marg[base + r] /
                                 (__expf(uvec[(size_t)b * I_ + base + r]) * sv);
                a_out[(size_t)b * I_ + base + r] = fminf(__powf(x, PCOEF_), HUGE_);
            }
        }
    }
}

// ---------------------------------------------------------------------------
// Col pass: s2_j = scale * sum_i K_ij * a_i ; optional b_j = (nu/(e^v s2))^p
// Each wave owns 16 cols x all 1024 rows.  K is the (row-major, coalesced)
// B operand; a is broadcast into A; every D row holds the 16 column sums.
__global__ __launch_bounds__(256)
void k_colpass(const float* __restrict__ K, const float* __restrict__ avec,
               float* __restrict__ s2_out, const float* __restrict__ marg,
               const float* __restrict__ vvec, float* __restrict__ bn_out,
               float scale, int compute_b)
{
    const int wave = threadIdx.x >> 5;
    const int lane = threadIdx.x & 31;
    const int b    = blockIdx.x >> 3;             // 8 col-strips per batch
    const int j0   = (blockIdx.x & 7) * 128 + wave * 16;
    const float* Kb = K + (size_t)b * I_ * J_;
    const float* aa = avec + (size_t)b * I_;
    const int n   = lane & 15;
    const int khi = lane >> 4;

    v8f acc = {};
    for (int i0 = 0; i0 < I_; i0 += 4) {
        v2f afrag;                                // A[m][k] = a[i0+k] (bcast rows)
        afrag.x = aa[i0 + 2 * khi];
        afrag.y = aa[i0 + 2 * khi + 1];
        v2f bfrag;                                // B[k][n] = K[i0+k][j0+n]
        bfrag.x = Kb[(size_t)(i0 + khi) * J_ + j0 + n];
        bfrag.y = Kb[(size_t)(i0 + 2 + khi) * J_ + j0 + n];
        acc = __builtin_amdgcn_wmma_f32_16x16x4_f32(false, afrag, false, bfrag,
                                                    (short)0, acc, false, false);
    }
    if (lane < 16) {                              // D row M=0: vgpr0, lanes 0-15
        const float sv = acc[0] * scale;
        s2_out[(size_t)b * J_ + j0 + lane] = sv;
        if (compute_b) {
            const float x = marg[j0 + lane] /
                            (__expf(vvec[(size_t)b * J_ + j0 + lane]) * sv);
            bn_out[(size_t)b * J_ + j0 + lane] = fminf(__powf(x, PCOEF_), HUGE_);
        }
    }
}

// ---------------------------------------------------------------------------
// Single-block: global max(a), max(bn) -> absorb flag; apply u/v/b updates.
__global__ __launch_bounds__(256)
void k_absorb(float* __restrict__ u, float* __restrict__ v,
              const float* __restrict__ a, const float* __restrict__ bn,
              float* __restrict__ bvec, int* __restrict__ flag, int iter)
{
    __shared__ float red[256];
    __shared__ int   sflag;
    const int t = threadIdx.x;
    float mA = 0.0f, mB = 0.0f;
    for (int idx = t; idx < B_ * I_; idx += 256) mA = fmaxf(mA, a[idx]);
    for (int idx = t; idx < B_ * J_; idx += 256) mB = fmaxf(mB, bn[idx]);
    red[t] = mA; __syncthreads();
    for (int o = 128; o > 0; o >>= 1) {
        if (t < o) red[t] = fmaxf(red[t], red[t + o]);
        __syncthreads();
    }
    const float maxa = red[0]; __syncthreads();
    red[t] = mB; __syncthreads();
    for (int o = 128; o > 0; o >>= 1) {
        if (t < o) red[t] = fmaxf(red[t], red[t + o]);
        __syncthreads();
    }
    const float maxb = red[0];
    if (t == 0) {
        sflag = (maxa > BIG_) || (maxb > BIG_) || (iter == NITER_ - 1);
        *flag = sflag;
    }
    __syncthreads();
    const int f = sflag;
    for (int idx = t; idx < B_ * I_; idx += 256) {
        if (f) {
            u[idx] += EPS_ * __logf(a[idx]);
            v[idx] += EPS_ * __logf(bn[idx]);
            bvec[idx] = 1.0f;
        } else {
            bvec[idx] = bn[idx];
        }
    }
}

// ---------------------------------------------------------------------------
// cons1/cons2: KL marginals.  16 blocks: [0..7]=rows vs mu, [8..15]=cols vs nu.
__global__ __launch_bounds__(256)
void k_cons(const float* __restrict__ rmu, const float* __restrict__ mu,
            const float* __restrict__ rnu, const float* __restrict__ nu,
            float* __restrict__ out)
{
    __shared__ float red[256];
    const int which = blockIdx.x >> 3;
    const int b     = blockIdx.x & 7;
    const float* r = which ? (rnu + (size_t)b * J_) : (rmu + (size_t)b * I_);
    const float* m = which ? nu : mu;
    float sum = 0.0f;
    for (int idx = threadIdx.x; idx < I_; idx += 256) {
        const float x = r[idx], y = m[idx];
        const float div = x / (y + SMALL_);
        sum += y * (div * __logf(div + SMALL_) - div + 1.0f);
    }
    red[threadIdx.x] = sum; __syncthreads();
    for (int o = 128; o > 0; o >>= 1) {
        if (threadIdx.x < o) red[threadIdx.x] += red[threadIdx.x + o];
        __syncthreads();
    }
    if (threadIdx.x == 0) out[8 + which * 8 + b] = red[0] * DX_;
}

// ---------------------------------------------------------------------------
// transport = sum(K*C)*dx*dy, two deterministic stages.
__global__ __launch_bounds__(256)
void k_tpart(const float* __restrict__ K, const float* __restrict__ C,
             float* __restrict__ partial)
{
    __shared__ float red[256];
    const int    b     = blockIdx.x >> 7;         // 128 chunks per batch
    const int    chunk = blockIdx.x & 127;
    const size_t base  = (size_t)b * I_ * J_ + (size_t)chunk * 8192;
    float sum = 0.0f;
    #pragma unroll 4
    for (int k = 0; k < 32; ++k) {
        const size_t idx = base + threadIdx.x + (size_t)k * 256;
        sum += K[idx] * C[idx];
    }
    red[threadIdx.x] = sum; __syncthreads();
    for (int o = 128; o > 0; o >>= 1) {
        if (threadIdx.x < o) red[threadIdx.x] += red[threadIdx.x + o];
        __syncthreads();
    }
    if (threadIdx.x == 0) partial[blockIdx.x] = red[0];
}

__global__ __launch_bounds__(128)
void k_tred(const float* __restrict__ partial, float* __restrict__ out)
{
    __shared__ float red[128];
    red[threadIdx.x] = partial[blockIdx.x * 128 + threadIdx.x];
    __syncthreads();
    for (int o = 64; o > 0; o >>= 1) {
        if (threadIdx.x < o) red[threadIdx.x] += red[threadIdx.x + o];
        __syncthreads();
    }
    if (threadIdx.x == 0) out[blockIdx.x] = red[0] * (DX_ * DY_);
}

// ---------------------------------------------------------------------------
extern "C" void kernel_launch(void* const* d_in, const int* in_sizes, int n_in,
                              void* d_out, int out_size, void* d_ws, size_t ws_size,
                              hipStream_t stream)
{
    const float* C  = (const float*)d_in[0];     // [8,1024,1024]
    const float* mu = (const float*)d_in[1];     // [1024]
    const float* nu = (const float*)d_in[2];     // [1024]
    float* out = (float*)d_out;                  // [8 transport][8 cons1][8 cons2][K]
    float* K   = out + 24;                       // K lives in the output buffer

    float* ws   = (float*)d_ws;                  // ~300 KB scratch
    float* u    = ws;                            // [B,I]
    float* v    = ws + 8192;                     // [B,J]
    float* a    = ws + 16384;
    float* bvec = ws + 24576;
    float* bn   = ws + 32768;
    float* s    = ws + 40960;
    float* s2   = ws + 49152;
    float* rmu  = ws + 57344;
    float* rnu  = ws + 65536;
    float* part = ws + 73728;                    // [1024]
    int*   flag = (int*)(ws + 74752);

    k_init<<<32, 256, 0, stream>>>(u, v, bvec);
    k_recompute<<<8192, 256, 0, stream>>>(C, u, v, K, flag, 1);

    for (int it = 0; it < NITER_; ++it) {
        k_rowpass<<<64, 256, 0, stream>>>(K, bvec, s, mu, u, a, DY_, 1);
        k_colpass<<<64, 256, 0, stream>>>(K, a, s2, nu, v, bn, DX_, 1);
        k_absorb<<<1, 256, 0, stream>>>(u, v, a, bn, bvec, flag, it);
        k_recompute<<<8192, 256, 0, stream>>>(C, u, v, K, flag, 0);
    }

    // last iteration forces absorption: bvec == 1, K is final.
    k_rowpass<<<64, 256, 0, stream>>>(K, bvec, rmu, mu, u, a, DY_, 0);
    k_colpass<<<64, 256, 0, stream>>>(K, bvec, rnu, nu, v, bn, DX_, 0);
    k_cons<<<16, 256, 0, stream>>>(rmu, mu, rnu, nu, out);
    k_tpart<<<1024, 256, 0, stream>>>(K, C, part);
    k_tred<<<8, 128, 0, stream>>>(part, out);
}